// label_loss_60644938220048
// MI455X (gfx1250) — compile-verified
//
#include <hip/hip_runtime.h>

// ---------------------------------------------------------------------------
// label_loss reduction for MI455X (gfx1250, wave32)
//   inputs : d_in[0] label   (B,3,2) f32
//            d_in[1] est     (B,6)   f32
//            d_in[2] mix_way (B,)    i32
//   output : d_out[0..2] = label_loss, mae1, mae2   (f32)
//   d_ws   : per-block partial sums (5 floats per block)
//
// Data path: CDNA5 async global->LDS copies (ASYNCcnt), double-buffered
// 256-pair tiles, per-wave s_wait_asynccnt only (each thread consumes only
// LDS slots its own wave filled -> no workgroup barriers needed).
// ---------------------------------------------------------------------------

static constexpr float RAD2DEG  = 57.295779513082320876798154814105f;
static constexpr int   NTHREADS = 256;   // 8 wave32 per block
static constexpr int   TILE     = 256;   // pairs per tile (== NTHREADS)

// ---- feature detection (device pass only) ---------------------------------
#if defined(__HIP_DEVICE_COMPILE__) && defined(__has_builtin)
#  if __has_builtin(__builtin_amdgcn_global_load_async_to_lds_b128) && \
      __has_builtin(__builtin_amdgcn_global_load_async_to_lds_b64)
#    define USE_ASYNC_LDS 1
#  else
#    define USE_ASYNC_LDS 0
#  endif
#else
#  define USE_ASYNC_LDS 0
#endif

#if USE_ASYNC_LDS
// Builtin signature (from hipcc diagnostics):
//   void __builtin_amdgcn_global_load_async_to_lds_b128(int4 AS1* gsrc,
//        int4 AS3* ldst, int imm_offset, int imm_cpol)
// The immediate offset applies to BOTH the global and the LDS address.
typedef __attribute__((ext_vector_type(4))) int v4i;
typedef __attribute__((ext_vector_type(2))) int v2i;
typedef __attribute__((address_space(1))) v4i* g_v4i_p;
typedef __attribute__((address_space(3))) v4i* l_v4i_p;
typedef __attribute__((address_space(1))) v2i* g_v2i_p;
typedef __attribute__((address_space(3))) v2i* l_v2i_p;

__device__ __forceinline__ void wait_async_le(int which /*0 or 7*/) {
#  if __has_builtin(__builtin_amdgcn_s_wait_asynccnt)
    if (which == 0) __builtin_amdgcn_s_wait_asynccnt(0);
    else            __builtin_amdgcn_s_wait_asynccnt(7);
#  else
    if (which == 0) asm volatile("s_wait_asynccnt 0x0" ::: "memory");
    else            asm volatile("s_wait_asynccnt 0x7" ::: "memory");
#  endif
}
#endif

__device__ __forceinline__ float wrap_deg(float x) {
    float a = fabsf(x);
    return a > 180.0f ? 360.0f - a : a;
}

__device__ __forceinline__ float mse3(float ax, float ay, float az,
                                      float bx, float by, float bz) {
    float dx = ax - bx, dy = ay - by, dz = az - bz;
    return (dx * dx + dy * dy + dz * dz) * (1.0f / 3.0f);
}

struct Acc { float s1, mx, c0, m1, m2; };

__device__ __forceinline__ void process_row(const float* __restrict__ l,
                                            const float* __restrict__ e,
                                            int mix, Acc& a) {
    const float l1x = l[0], l1y = l[2], l1z = l[4];
    const float l2x = l[1], l2y = l[3], l2z = l[5];
    const float e1x = e[0], e1y = e[1], e1z = e[2];
    const float e2x = e[3], e2y = e[4], e2z = e[5];

    const float m11 = mse3(e1x, e1y, e1z, l1x, l1y, l1z);
    const float m22 = mse3(e2x, e2y, e2z, l2x, l2y, l2z);
    const float m21 = mse3(e2x, e2y, e2z, l1x, l1y, l1z);
    const float m12 = mse3(e1x, e1y, e1z, l2x, l2y, l2z);
    const float mixed = fminf(m11 + m22, m21 + m12);

    a.s1 += m11;                       // feeds global single_loss
    if (mix != 0) a.mx += mixed;       // mixed branch contribution
    else          a.c0 += 1.0f;        // count of single_loss rows

    const float laz1 = atan2f(l1y, l1x) * RAD2DEG;
    const float laz2 = atan2f(l2y, l2x) * RAD2DEG;
    const float eaz1 = atan2f(e1y, e1x) * RAD2DEG;
    const float eaz2 = atan2f(e2y, e2x) * RAD2DEG;

    const float e11 = wrap_deg(laz1 - eaz1);
    const float e22 = wrap_deg(laz2 - eaz2);
    const float e12 = wrap_deg(laz1 - eaz2);
    const float e21 = wrap_deg(laz2 - eaz1);

    const bool direct = (e11 + e22) < (e12 + e21);
    a.m1 += direct ? e11 : e12;
    a.m2 += direct ? e22 : e21;
}

__device__ __forceinline__ void process_f4(const float4 la, const float4 lb,
                                           const float4 lc, const float4 ea,
                                           const float4 eb, const float4 ec,
                                           const int2 mw, Acc& a) {
    const float lbuf[12] = {la.x, la.y, la.z, la.w, lb.x, lb.y, lb.z, lb.w,
                            lc.x, lc.y, lc.z, lc.w};
    const float ebuf[12] = {ea.x, ea.y, ea.z, ea.w, eb.x, eb.y, eb.z, eb.w,
                            ec.x, ec.y, ec.z, ec.w};
    process_row(lbuf,     ebuf,     mw.x, a);
    process_row(lbuf + 6, ebuf + 6, mw.y, a);
}

// direct-load fallback / tail path
__device__ __forceinline__ void process_pair_global(size_t i,
                                                    const float4* __restrict__ L4,
                                                    const float4* __restrict__ E4,
                                                    const int2* __restrict__ M2,
                                                    Acc& a) {
    const size_t base = 3u * i;
    process_f4(L4[base], L4[base + 1], L4[base + 2],
               E4[base], E4[base + 1], E4[base + 2], M2[i], a);
}

__device__ __forceinline__ void block_reduce_store(Acc a, float* __restrict__ dst) {
    #pragma unroll
    for (int off = 16; off > 0; off >>= 1) {
        a.s1 += __shfl_down(a.s1, off, 32);
        a.mx += __shfl_down(a.mx, off, 32);
        a.c0 += __shfl_down(a.c0, off, 32);
        a.m1 += __shfl_down(a.m1, off, 32);
        a.m2 += __shfl_down(a.m2, off, 32);
    }
    __shared__ float sm[NTHREADS / 32][5];
    const int wave = threadIdx.x >> 5;
    const int lane = threadIdx.x & 31;
    if (lane == 0) {
        sm[wave][0] = a.s1; sm[wave][1] = a.mx; sm[wave][2] = a.c0;
        sm[wave][3] = a.m1; sm[wave][4] = a.m2;
    }
    __syncthreads();
    if (threadIdx.x == 0) {
        float r0 = 0.f, r1 = 0.f, r2 = 0.f, r3 = 0.f, r4 = 0.f;
        #pragma unroll
        for (int w = 0; w < NTHREADS / 32; ++w) {
            r0 += sm[w][0]; r1 += sm[w][1]; r2 += sm[w][2];
            r3 += sm[w][3]; r4 += sm[w][4];
        }
        dst[0] = r0; dst[1] = r1; dst[2] = r2; dst[3] = r3; dst[4] = r4;
    }
}

__global__ void label_loss_partial_kernel(const float* __restrict__ label,
                                          const float* __restrict__ est,
                                          const int*   __restrict__ mix,
                                          float* __restrict__ partials,
                                          int npairs) {
    const float4* __restrict__ L4 = reinterpret_cast<const float4*>(label);
    const float4* __restrict__ E4 = reinterpret_cast<const float4*>(est);
    const int2*   __restrict__ M2 = reinterpret_cast<const int2*>(mix);

    Acc a{0.f, 0.f, 0.f, 0.f, 0.f};
    const int tid = threadIdx.x;

#if USE_ASYNC_LDS
    // double-buffered LDS tiles: 2 * (12KB label + 12KB est + 2KB mix) = 52KB
    __shared__ float4 s_lab[2][TILE * 3];
    __shared__ float4 s_est[2][TILE * 3];
    __shared__ int2   s_mix[2][TILE];

    const int ntiles = npairs / TILE;      // full tiles (npairs = 2^21 -> exact)

    auto issue_tile = [&](int tile, int buf) {
        const size_t p = (size_t)tile * TILE + tid;
        g_v4i_p gl = (g_v4i_p)(L4 + 3 * p);
        l_v4i_p ll = (l_v4i_p)&s_lab[buf][tid * 3];
        g_v4i_p ge = (g_v4i_p)(E4 + 3 * p);
        l_v4i_p le = (l_v4i_p)&s_est[buf][tid * 3];
        g_v2i_p gm = (g_v2i_p)(M2 + p);
        l_v2i_p lm = (l_v2i_p)&s_mix[buf][tid];
        __builtin_amdgcn_global_load_async_to_lds_b128(gl, ll, 0,  0);
        __builtin_amdgcn_global_load_async_to_lds_b128(gl, ll, 16, 0);
        __builtin_amdgcn_global_load_async_to_lds_b128(gl, ll, 32, 0);
        __builtin_amdgcn_global_load_async_to_lds_b128(ge, le, 0,  0);
        __builtin_amdgcn_global_load_async_to_lds_b128(ge, le, 16, 0);
        __builtin_amdgcn_global_load_async_to_lds_b128(ge, le, 32, 0);
        __builtin_amdgcn_global_load_async_to_lds_b64 (gm, lm, 0,  0);
    };

    int t = blockIdx.x;
    int buf = 0;
    if (t < ntiles) issue_tile(t, buf);
    for (; t < ntiles; t += gridDim.x) {
        const int nt = t + (int)gridDim.x;
        const bool have_next = nt < ntiles;
        if (have_next) issue_tile(nt, buf ^ 1);
        wait_async_le(have_next ? 7 : 0);  // current tile's 7 ops retired
        process_f4(s_lab[buf][tid * 3], s_lab[buf][tid * 3 + 1],
                   s_lab[buf][tid * 3 + 2],
                   s_est[buf][tid * 3], s_est[buf][tid * 3 + 1],
                   s_est[buf][tid * 3 + 2],
                   s_mix[buf][tid], a);
        buf ^= 1;
    }
    // tail pairs (none for B = 4194304, defensive): block 0 direct loads
    if (blockIdx.x == 0) {
        for (int i = ntiles * TILE + tid; i < npairs; i += NTHREADS)
            process_pair_global((size_t)i, L4, E4, M2, a);
    }
#else
    const int gtid   = blockIdx.x * blockDim.x + tid;
    const int stride = gridDim.x * blockDim.x;
    for (int i = gtid; i < npairs; i += stride) {
        if (i + stride < npairs) {
            __builtin_prefetch(&L4[3u * (size_t)(i + stride)], 0, 3);
            __builtin_prefetch(&E4[3u * (size_t)(i + stride)], 0, 3);
        }
        process_pair_global((size_t)i, L4, E4, M2, a);
    }
#endif

    block_reduce_store(a, partials + (size_t)blockIdx.x * 5);
}

__global__ void label_loss_finalize_kernel(const float* __restrict__ partials,
                                           int nblocks, int btotal,
                                           float* __restrict__ out) {
    Acc a{0.f, 0.f, 0.f, 0.f, 0.f};
    for (int i = threadIdx.x; i < nblocks; i += blockDim.x) {
        const float* p = partials + (size_t)i * 5;
        a.s1 += p[0]; a.mx += p[1]; a.c0 += p[2]; a.m1 += p[3]; a.m2 += p[4];
    }
    __shared__ float res[5];
    block_reduce_store(a, res);
    __syncthreads();
    if (threadIdx.x == 0) {
        const float invB = 1.0f / (float)btotal;
        const float single_loss = res[0] * invB;                 // S1 / B
        out[0] = (res[2] * single_loss + res[1]) * invB;         // (C0*single + M)/B
        out[1] = res[3] * invB;                                  // mae1
        out[2] = res[4] * invB;                                  // mae2
    }
}

extern "C" void kernel_launch(void* const* d_in, const int* in_sizes, int n_in,
                              void* d_out, int out_size, void* d_ws, size_t ws_size,
                              hipStream_t stream) {
    const float* label = (const float*)d_in[0];
    const float* est   = (const float*)d_in[1];
    const int*   mix   = (const int*)d_in[2];
    float* out      = (float*)d_out;
    float* partials = (float*)d_ws;

    const int btotal = in_sizes[0] / 6;   // label is (B,3,2)
    const int npairs = btotal / 2;        // B = 4194304 is even

    int nblocks = 2048;                   // 2048 blocks x 256 thr -> 4 tiles each
    const size_t need = (size_t)nblocks * 5 * sizeof(float);
    if (need > ws_size) {
        nblocks = (int)(ws_size / (5 * sizeof(float)));
        if (nblocks < 1) nblocks = 1;
        if (nblocks > 2048) nblocks = 2048;
    }

    label_loss_partial_kernel<<<nblocks, NTHREADS, 0, stream>>>(
        label, est, mix, partials, npairs);
    label_loss_finalize_kernel<<<1, NTHREADS, 0, stream>>>(
        partials, nblocks, btotal, out);
}